// MultiHeadAttention_10033043603536
// MI455X (gfx1250) — compile-verified
//
#include <hip/hip_runtime.h>
#include <hip/hip_bf16.h>

// ---------------------------------------------------------------------------
// Multi-head attention forward for MI455X (gfx1250), fp32 WMMA path.
// B=2, S=2048, D=1024, H=16, DK=64. Memory-bound problem (~50MB HBM traffic,
// ~70 GFLOP) -> full-precision V_WMMA_F32_16X16X4_F32 everywhere.
// Projections: LDS-staged weight panels via global_load_async_to_lds_b128
// (ASYNCcnt) + per-wave 16x64 register blocking for 4x A-fragment reuse.
// ---------------------------------------------------------------------------

typedef __attribute__((ext_vector_type(2))) float v2f;
typedef __attribute__((ext_vector_type(8))) float v8f;

#define WMMA_F32(a, b, c) \
    __builtin_amdgcn_wmma_f32_16x16x4_f32(false, (a), false, (b), (short)0, (c), false, false)

namespace cfg {
constexpr int B  = 2;
constexpr int S  = 2048;
constexpr int D  = 1024;
constexpr int H  = 16;
constexpr int DK = 64;          // D / H
constexpr int M  = B * S;       // 4096 rows total
constexpr int KC = 32;          // K-panel depth staged in LDS
constexpr int NC = 256;         // N-panel width staged in LDS
}

__device__ __forceinline__ v8f vzero8() {
    v8f v;
#pragma unroll
    for (int i = 0; i < 8; ++i) v[i] = 0.0f;
    return v;
}

// Async global->LDS copy, 16B per lane, tracked with ASYNCcnt.
__device__ __forceinline__ void async_load_b128(unsigned lds_off, const void* gptr) {
    asm volatile("global_load_async_to_lds_b128 %0, %1, off"
                 :: "v"(lds_off), "v"(gptr) : "memory");
}
__device__ __forceinline__ void wait_asynccnt0() {
    asm volatile("s_wait_asynccnt 0x0" ::: "memory");
}

// ---------------------------------------------------------------------------
// Shared GEMM core: C[16x64 per wave] += A[M,K] * W[K,N-panel].
// Block = 256 threads (8 waves) computing a 32x256 tile; the 32xNC weight
// K-panel is staged in LDS by async copies and consumed by all 8 waves.
// All waves have identical trip counts -> __syncthreads is convergent.
// ---------------------------------------------------------------------------
__device__ __forceinline__ void panel_gemm(
    const float* __restrict__ Arow,     // A + (row0+ln)*D + 2*half
    const float* __restrict__ Wn0,      // W + n0 (column panel base)
    float* __restrict__ Wlds,           // [KC * NC] floats in LDS
    int tid, int half, int ln, int nw,  // nw = wave&3 (N sub-tile)
    v8f acc[4])
{
    using namespace cfg;
    const unsigned lds_base = (unsigned)(uintptr_t)&Wlds[0];

    for (int k0 = 0; k0 < D; k0 += KC) {
        // ---- stage W[k0..k0+KC) x [n0..n0+NC) into LDS (KC*NC*4 = 32KB) ----
#pragma unroll
        for (int j = 0; j < (KC * NC / 4) / 256; ++j) {   // 8 x b128 per thread
            const int idx = j * 256 + tid;                // float4 index
            const int r   = idx >> 6;                     // panel row 0..31
            const int c4  = idx & 63;                     // float4 column
            async_load_b128(lds_base + (unsigned)idx * 16,
                            Wn0 + (size_t)(k0 + r) * D + c4 * 4);
        }
        // prefetch next A panel while the async copy is in flight
        if (k0 + KC < D) __builtin_prefetch(Arow + k0 + KC, 0, 0);
        wait_asynccnt0();
        __syncthreads();

        // ---- compute: 8 K-steps x 4 N-tiles of WMMA ----
#pragma unroll
        for (int kk = 0; kk < KC; kk += 4) {
            v2f a;
            a.x = Arow[k0 + kk];          // Arow already offset by 2*half
            a.y = Arow[k0 + kk + 1];
            const float* wrow = &Wlds[(kk + 2 * half) * NC + nw * 64 + ln];
#pragma unroll
            for (int d = 0; d < 4; ++d) {
                v2f b;
                b.x = wrow[16 * d];
                b.y = wrow[NC + 16 * d];
                acc[d] = WMMA_F32(a, b, acc[d]);
            }
        }
        __syncthreads();
    }
}

// ---------------------------------------------------------------------------
// Kernel 1: fused Q/K/V projection.  grid = (M/32, D/NC, 3), block = 256.
// Output scattered into [B, H, S, DK] layout for the attention kernel.
// ---------------------------------------------------------------------------
__global__ __launch_bounds__(256) void qkv_proj_kernel(
    const float* __restrict__ x,
    const float* __restrict__ Wq, const float* __restrict__ bq,
    const float* __restrict__ Wk, const float* __restrict__ bk,
    const float* __restrict__ Wv, const float* __restrict__ bv,
    float* __restrict__ Qw, float* __restrict__ Kw, float* __restrict__ Vw)
{
    using namespace cfg;
    __shared__ float Wlds[KC * NC];

    const int tid  = threadIdx.x;
    const int lane = tid & 31;
    const int wave = tid >> 5;
    const int half = lane >> 4;
    const int ln   = lane & 15;
    const int nw   = wave & 3;

    const float* W;
    const float* bias;
    float* Out;
    if (blockIdx.z == 0)      { W = Wq; bias = bq; Out = Qw; }
    else if (blockIdx.z == 1) { W = Wk; bias = bk; Out = Kw; }
    else                      { W = Wv; bias = bv; Out = Vw; }

    const int row0 = blockIdx.x * 32 + (wave >> 2) * 16;
    const int n0   = blockIdx.y * NC;
    const int col0 = n0 + nw * 64;

    v8f acc[4];
#pragma unroll
    for (int d = 0; d < 4; ++d) acc[d] = vzero8();

    const float* Arow = x + (size_t)(row0 + ln) * D + 2 * half;
    panel_gemm(Arow, W + n0, Wlds, tid, half, ln, nw, acc);

    // Scatter to [B,H,S,DK]; h is constant per wave (col0 is 64-aligned).
    const int h = col0 >> 6;
#pragma unroll
    for (int d = 0; d < 4; ++d) {
        const int gc = col0 + 16 * d + ln;
        const int dk = gc & (DK - 1);
        const float bval = bias[gc];
#pragma unroll
        for (int r = 0; r < 8; ++r) {
            const int gr = row0 + r + 8 * half;
            const int bb = gr >> 11;       // / S
            const int s  = gr & (S - 1);
            Out[(((size_t)bb * H + h) * S + s) * DK + dk] = acc[d][r] + bval;
        }
    }
}

// ---------------------------------------------------------------------------
// Kernel 2: causal flash attention.  One wave per (b, h, 16-row query tile).
// grid = B*H*(S/16)/8 = 512 blocks of 256 threads (8 waves).
// Online softmax with width-16 shuffle reductions; P re-layout via LDS with
// an explicit wave-level s_wait_dscnt fence (no workgroup barrier -> no
// divergent-barrier hazard across waves with different causal trip counts).
// ---------------------------------------------------------------------------
__global__ __launch_bounds__(256) void attn_kernel(
    const float* __restrict__ Qw, const float* __restrict__ Kw,
    const float* __restrict__ Vw, float* __restrict__ ctx)
{
    using namespace cfg;
    __shared__ float Plds[8 * 256];   // 16x16 P tile per wave

    const int lane = threadIdx.x & 31;
    const int wave = threadIdx.x >> 5;
    const int half = lane >> 4;
    const int ln   = lane & 15;

    const int gw = blockIdx.x * 8 + wave;     // global wave id
    const int qt = gw & 127;                  // S/16 = 128 query tiles
    const int h  = (gw >> 7) & (H - 1);
    const int b  = gw >> 11;
    const int q0 = qt * 16;

    const size_t headbase = ((size_t)b * H + h) * S * DK;
    const float* Qb = Qw + headbase;
    const float* Kb = Kw + headbase;
    const float* Vb = Vw + headbase;
    float* Pw = &Plds[wave * 256];

    // Preload the Q tile as 16 A-fragments (reused across all key tiles).
    v2f aq[16];
    {
        const float* qrow = Qb + (size_t)(q0 + ln) * DK + 2 * half;
#pragma unroll
        for (int t = 0; t < 16; ++t) {
            aq[t].x = qrow[4 * t];
            aq[t].y = qrow[4 * t + 1];
        }
    }

    v8f cacc[4];
#pragma unroll
    for (int d = 0; d < 4; ++d) cacc[d] = vzero8();

    float mrow[8], lrow[8];
#pragma unroll
    for (int r = 0; r < 8; ++r) { mrow[r] = -1e30f; lrow[r] = 0.0f; }

    const float scale = 0.125f;   // 1/sqrt(DK)
    const int nkt = qt + 1;       // causal: key tiles 0..qt

    for (int kt = 0; kt < nkt; ++kt) {
        const int k0 = kt * 16;

        // ---- scores S = Q K^T  (16 WMMA steps over DK=64) ----
        v8f sacc = vzero8();
        {
            const float* krow = Kb + (size_t)(k0 + ln) * DK + 2 * half;
#pragma unroll
            for (int t = 0; t < 16; ++t) {
                v2f bk;
                bk.x = krow[4 * t];
                bk.y = krow[4 * t + 1];
                sacc = WMMA_F32(aq[t], bk, sacc);
            }
        }

        const bool diag = (kt == qt);
        float p[8];
#pragma unroll
        for (int r = 0; r < 8; ++r) {
            float sv = sacc[r] * scale;
            // causal mask on diagonal tile: col (k0+ln) > row (q0+r+8*half)
            if (diag && (k0 + ln > q0 + r + 8 * half)) sv = -1e9f;

            // row max across the 16-lane group holding this row
            float t = sv;
            t = fmaxf(t, __shfl_xor(t, 1, 16));
            t = fmaxf(t, __shfl_xor(t, 2, 16));
            t = fmaxf(t, __shfl_xor(t, 4, 16));
            t = fmaxf(t, __shfl_xor(t, 8, 16));
            const float mn = fmaxf(mrow[r], t);
            const float al = __expf(mrow[r] - mn);
            const float pv = __expf(sv - mn);

            float rs = pv;
            rs += __shfl_xor(rs, 1, 16);
            rs += __shfl_xor(rs, 2, 16);
            rs += __shfl_xor(rs, 4, 16);
            rs += __shfl_xor(rs, 8, 16);

            lrow[r] = lrow[r] * al + rs;
            mrow[r] = mn;
            p[r]    = pv;
#pragma unroll
            for (int d = 0; d < 4; ++d) cacc[d][r] *= al;
        }

        // ---- re-layout P (C layout -> A layout) through LDS ----
#pragma unroll
        for (int r = 0; r < 8; ++r)
            Pw[(r + 8 * half) * 16 + ln] = p[r];
        asm volatile("s_wait_dscnt 0x0" ::: "memory");   // wave-level LDS fence

        // ---- ctx += P V  (4 K-steps x 4 DK tiles) ----
#pragma unroll
        for (int t = 0; t < 4; ++t) {
            v2f ap;
            ap.x = Pw[ln * 16 + 4 * t + 2 * half];
            ap.y = Pw[ln * 16 + 4 * t + 2 * half + 1];
            const float* vrow = Vb + (size_t)(k0 + 4 * t + 2 * half) * DK + ln;
#pragma unroll
            for (int d = 0; d < 4; ++d) {
                v2f bv;
                bv.x = vrow[16 * d];
                bv.y = vrow[DK + 16 * d];
                cacc[d] = WMMA_F32(ap, bv, cacc[d]);
            }
        }
    }

    // ---- normalize and store ctx tile back in [B, S, D] layout ----
#pragma unroll
    for (int r = 0; r < 8; ++r) {
        const float inv = 1.0f / lrow[r];
        const int srow = q0 + r + 8 * half;
        float* orow = ctx + ((size_t)b * S + srow) * D + h * DK + ln;
#pragma unroll
        for (int d = 0; d < 4; ++d)
            orow[16 * d] = cacc[d][r] * inv;
    }
}

// ---------------------------------------------------------------------------
// Kernel 3: output projection  out = ctx @ Wo + bo.
// grid = (M/32, D/NC), block = 256.
// ---------------------------------------------------------------------------
__global__ __launch_bounds__(256) void out_proj_kernel(
    const float* __restrict__ ctx, const float* __restrict__ Wo,
    const float* __restrict__ bo, float* __restrict__ out)
{
    using namespace cfg;
    __shared__ float Wlds[KC * NC];

    const int tid  = threadIdx.x;
    const int lane = tid & 31;
    const int wave = tid >> 5;
    const int half = lane >> 4;
    const int ln   = lane & 15;
    const int nw   = wave & 3;

    const int row0 = blockIdx.x * 32 + (wave >> 2) * 16;
    const int n0   = blockIdx.y * NC;
    const int col0 = n0 + nw * 64;

    v8f acc[4];
#pragma unroll
    for (int d = 0; d < 4; ++d) acc[d] = vzero8();

    const float* Arow = ctx + (size_t)(row0 + ln) * D + 2 * half;
    panel_gemm(Arow, Wo + n0, Wlds, tid, half, ln, nw, acc);

#pragma unroll
    for (int d = 0; d < 4; ++d) {
        const int gc = col0 + 16 * d + ln;
        const float bval = bo[gc];
#pragma unroll
        for (int r = 0; r < 8; ++r) {
            const int gr = row0 + r + 8 * half;
            out[(size_t)gr * D + gc] = acc[d][r] + bval;
        }
    }
}

// ---------------------------------------------------------------------------
// Host-side launch.  Inputs (setup_inputs order):
//   0:x 1:mask 2:Wq 3:bq 4:Wk 5:bk 6:Wv 7:bv 8:Wo 9:bo
// Workspace: Q (16MB) | K (16MB) | V (16MB) | ctx (16MB)  = 64MB.
// ---------------------------------------------------------------------------
extern "C" void kernel_launch(void* const* d_in, const int* in_sizes, int n_in,
                              void* d_out, int out_size, void* d_ws, size_t ws_size,
                              hipStream_t stream) {
    using namespace cfg;
    (void)in_sizes; (void)n_in; (void)out_size; (void)ws_size;

    const float* x  = (const float*)d_in[0];
    const float* Wq = (const float*)d_in[2];
    const float* bq = (const float*)d_in[3];
    const float* Wk = (const float*)d_in[4];
    const float* bk = (const float*)d_in[5];
    const float* Wv = (const float*)d_in[6];
    const float* bv = (const float*)d_in[7];
    const float* Wo = (const float*)d_in[8];
    const float* bo = (const float*)d_in[9];
    float* out = (float*)d_out;

    const size_t elems = (size_t)B * H * S * DK;   // 4M floats per tensor
    float* Qw   = (float*)d_ws;
    float* Kw   = Qw + elems;
    float* Vw   = Kw + elems;
    float* ctxb = Vw + elems;

    dim3 gProj(M / 32, D / NC, 3);
    qkv_proj_kernel<<<gProj, 256, 0, stream>>>(x, Wq, bq, Wk, bk, Wv, bv, Qw, Kw, Vw);

    const int nwaves = B * H * (S / 16);           // 4096 waves
    attn_kernel<<<nwaves / 8, 256, 0, stream>>>(Qw, Kw, Vw, ctxb);

    dim3 gOut(M / 32, D / NC, 1);
    out_proj_kernel<<<gOut, 256, 0, stream>>>(ctxb, Wo, bo, out);
}